// LeeOscillator_15839839387896
// MI455X (gfx1250) — compile-verified
//
#include <hip/hip_runtime.h>
#include <math.h>

// ---------------------------------------------------------------------------
// Lee oscillator, time-max-pooled over 100 steps. Purely elementwise and
// compute-bound (~1.1G FMA + 0.21G tanh vs 8MiB of traffic), so the CDNA5
// feature we target is the gfx1250 hardware transcendental V_TANH_F32
// (new on CDNA5; CDNA4 had no tanh TRANS op), co-executing with the FMA chain.
// 4 independent oscillator chains per thread provide the ILP to cover TRANS
// latency and enable B128 global accesses.
// ---------------------------------------------------------------------------

#define LEE_STEPS 100
#define LEE_VEC   4
#define LEE_BLOCK 256

// (a1,a2,a3,a4, b1,b2,b3,b4, xiE,xiI, mu, e, k) for oscillator types 1..8
__constant__ float c_par[8][13] = {
  { 0.0f, 5.0f,  5.0f,  1.0f,  0.0f, -1.0f,  1.0f,  0.0f, 0.0f, 0.0f, 5.0f, 0.001f, 500.0f},
  { 0.5f, 0.55f, 0.55f,-0.5f,  0.5f, -0.55f,-0.55f,-0.5f, 0.0f, 0.0f, 1.0f, 0.001f,  50.0f},
  { 0.5f, 0.6f,  0.55f, 0.5f, -0.5f, -0.6f, -0.55f, 0.5f, 0.0f, 0.0f, 1.0f, 0.001f,  50.0f},
  {-0.5f, 0.55f, 0.55f,-0.5f, -0.5f, -0.55f,-0.55f, 0.5f, 0.0f, 0.0f, 1.0f, 0.001f,  50.0f},
  {-0.9f, 0.9f,  0.9f, -0.9f,  0.9f, -0.9f, -0.9f,  0.9f, 0.0f, 0.0f, 1.0f, 0.001f,  50.0f},
  {-0.9f, 0.9f,  0.9f, -0.9f,  0.9f, -0.9f, -0.9f,  0.9f, 0.0f, 0.0f, 1.0f, 0.001f, 300.0f},
  {-5.0f, 5.0f,  5.0f, -5.0f,  1.0f, -1.0f, -1.0f,  1.0f, 0.0f, 0.0f, 1.0f, 0.001f,  50.0f},
  {-5.0f, 5.0f,  5.0f, -5.0f,  1.0f, -1.0f, -1.0f,  1.0f, 0.0f, 0.0f, 1.0f, 0.001f, 300.0f},
};

// gfx1250 hardware tanh (TRANS pipe). Builtin name guarded; inline-asm
// fallback guarantees v_tanh_f32 is emitted either way.
__device__ __forceinline__ float hw_tanh(float x) {
#if __has_builtin(__builtin_amdgcn_tanhf)
  return __builtin_amdgcn_tanhf(x);
#elif __has_builtin(__builtin_amdgcn_tanh_f32)
  return __builtin_amdgcn_tanh_f32(x);
#else
  float r;
  asm("v_tanh_f32 %0, %1" : "=v"(r) : "v"(x));
  return r;
#endif
}

__device__ __forceinline__ float hw_exp2(float x) {
#if __has_builtin(__builtin_amdgcn_exp2f)
  return __builtin_amdgcn_exp2f(x);
#else
  return exp2f(x);
#endif
}

__global__ __launch_bounds__(LEE_BLOCK) void lee_osc_kernel(
    const float* __restrict__ x, const int* __restrict__ osc_type,
    float* __restrict__ out, int n) {
  // --- uniform parameter fetch (device scalar) -----------------------------
  int t = *osc_type;
  t = (t < 1) ? 1 : ((t > 8) ? 8 : t);
  const float* p = c_par[t - 1];
  const float mu = p[10], ee = p[11], kk = p[12];
  // Fold mu into the affine coefficients: pre = cL*L + cE*E + cI*I + cS*s + c0
  const float eL = mu * p[0], eE = mu * p[1], eI = -mu * p[2], eS = mu * p[3], e0 = -mu * p[8];
  const float iL = mu * p[4], iE = -mu * p[5], iI = -mu * p[6], iS = mu * p[7], i0 = -mu * p[9];
  const float nk_log2e = -kk * 1.4426950408889634f; // exp(-k s^2) = exp2(nk_log2e*s^2)

  const int i0x = (blockIdx.x * LEE_BLOCK + threadIdx.x) * LEE_VEC;
  if (i0x >= n) return;
  const bool full = (i0x + LEE_VEC) <= n;

  // --- load 4 independent elements (B128 when full) ------------------------
  float xv[LEE_VEC];
  if (full) {
    const float4 v = *reinterpret_cast<const float4*>(x + i0x);
    xv[0] = v.x; xv[1] = v.y; xv[2] = v.z; xv[3] = v.w;
  } else {
    #pragma unroll
    for (int j = 0; j < LEE_VEC; ++j) xv[j] = (i0x + j < n) ? x[i0x + j] : 0.0f;
  }

  // --- per-element time-invariant terms ------------------------------------
  float s[LEE_VEC], damp[LEE_VEC], omega[LEE_VEC];
  float E[LEE_VEC], I[LEE_VEC], L[LEE_VEC], m[LEE_VEC];
  #pragma unroll
  for (int j = 0; j < LEE_VEC; ++j) {
    const float xj = xv[j];
    const float sg = (xj > 0.0f) ? 1.0f : ((xj < 0.0f) ? -1.0f : 0.0f);
    s[j]     = __builtin_fmaf(ee, sg, xj);           // sim0
    damp[j]  = hw_exp2(nk_log2e * s[j] * s[j]);      // exp(-k*sim0^2)
    omega[j] = hw_tanh(mu * s[j]);                   // tanh(mu*sim0)
    E[j] = 0.2f; I[j] = 0.0f; L[j] = 0.2f;
    m[j] = -__builtin_inff();
  }

  // --- 100-step recurrence: 4 independent chains give TRANS/VALU overlap ---
  #pragma unroll 2
  for (int step = 0; step < LEE_STEPS; ++step) {
    #pragma unroll
    for (int j = 0; j < LEE_VEC; ++j) {
      float pe = __builtin_fmaf(eS, s[j], e0);
      pe = __builtin_fmaf(eL, L[j], pe);
      pe = __builtin_fmaf(eE, E[j], pe);
      pe = __builtin_fmaf(eI, I[j], pe);
      float pi = __builtin_fmaf(iS, s[j], i0);
      pi = __builtin_fmaf(iL, L[j], pi);
      pi = __builtin_fmaf(iE, E[j], pi);
      pi = __builtin_fmaf(iI, I[j], pi);
      const float E1 = hw_tanh(pe);                  // v_tanh_f32
      const float I1 = hw_tanh(pi);                  // v_tanh_f32
      const float L1 = __builtin_fmaf(E1 - I1, damp[j], omega[j]);
      E[j] = E1; I[j] = I1; L[j] = L1;
      m[j] = fmaxf(m[j], L1);
    }
  }

  // --- store ---------------------------------------------------------------
  if (full) {
    float4 v; v.x = m[0]; v.y = m[1]; v.z = m[2]; v.w = m[3];
    *reinterpret_cast<float4*>(out + i0x) = v;       // global_store_b128
  } else {
    #pragma unroll
    for (int j = 0; j < LEE_VEC; ++j)
      if (i0x + j < n) out[i0x + j] = m[j];
  }
}

extern "C" void kernel_launch(void* const* d_in, const int* in_sizes, int n_in,
                              void* d_out, int out_size, void* d_ws, size_t ws_size,
                              hipStream_t stream) {
  (void)n_in; (void)out_size; (void)d_ws; (void)ws_size;
  const float* x        = (const float*)d_in[0];
  const int*   osc_type = (const int*)d_in[1];
  float*       out      = (float*)d_out;
  const int n = in_sizes[0];
  const int elems_per_block = LEE_BLOCK * LEE_VEC;
  const int grid = (n + elems_per_block - 1) / elems_per_block;
  lee_osc_kernel<<<grid, LEE_BLOCK, 0, stream>>>(x, osc_type, out, n);
}